// RBFLayer_71038759076230
// MI455X (gfx1250) — compile-verified
//
#include <hip/hip_runtime.h>

typedef float v2f __attribute__((ext_vector_type(2)));
typedef float v8f __attribute__((ext_vector_type(8)));

#define RBF_D 16
#define RBF_K 64

// One wave (32 lanes) handles 16 points x 64 centers.
// O[m][k] = ||Hx_m||^2 + ||Cs_k||^2 - 2 * (Hx_m . Cs_k), dot via V_WMMA_F32_16X16X4_F32.
__global__ __launch_bounds__(256)
void rbf_wmma_kernel(const float* __restrict__ x,
                     const float* __restrict__ centers,
                     const float* __restrict__ scaling,
                     float* __restrict__ O,
                     float* __restrict__ Hx) {
  const int lane = threadIdx.x & 31;
  const int wave = threadIdx.x >> 5;
  const int m    = lane & 15;   // row within 16-point / 16-center tile
  const int half = lane >> 4;   // which K-pair of the WMMA A/B striping

  const int point_base = (blockIdx.x * 8 + wave) * 16;
  const int row = point_base + m;

  // Per-lane scale values for d = 4c + 2*half + {0,1}  (tiny, cached)
  v2f sv[4];
#pragma unroll
  for (int c = 0; c < 4; ++c)
    sv[c] = *(const v2f*)(scaling + 4 * c + 2 * half);

  // Load this lane's half of point row `m`, scale -> A fragments; write Hx; partial ||h||^2.
  v2f a[4];
  float hsq_part = 0.f;
#pragma unroll
  for (int c = 0; c < 4; ++c) {
    const int d = 4 * c + 2 * half;
    v2f xv = *(const v2f*)(x + (size_t)row * RBF_D + d);
    v2f h;
    h.x = xv.x * sv[c].x;
    h.y = xv.y * sv[c].y;
    a[c] = h;
    hsq_part += h.x * h.x + h.y * h.y;
    *(v2f*)(Hx + (size_t)row * RBF_D + d) = h;  // Hx output, each element written once
  }
  // Combine the two lane-halves (each holds 8 of 16 dims of row m).
  const float hsq = hsq_part + __shfl_xor(hsq_part, 16, 32);
  // hval[v] = ||Hx||^2 of output row M = v + 8*half (C/D VGPR layout).
  float hval[8];
#pragma unroll
  for (int v = 0; v < 8; ++v)
    hval[v] = __shfl(hsq, v + 8 * half, 32);

  // 4 N-tiles of 16 centers cover K=64.
#pragma unroll
  for (int t = 0; t < 4; ++t) {
    const int col = t * 16 + m;
    v2f bmat[4];
    float csq_part = 0.f;
#pragma unroll
    for (int c = 0; c < 4; ++c) {
      const int d = 4 * c + 2 * half;
      v2f cv = *(const v2f*)(centers + (size_t)col * RBF_D + d);
      v2f bc;
      bc.x = cv.x * sv[c].x;
      bc.y = cv.y * sv[c].y;
      bmat[c] = bc;
      csq_part += bc.x * bc.x + bc.y * bc.y;
    }
    const float csq = csq_part + __shfl_xor(csq_part, 16, 32);

    // dot[m][col] = sum_d Hx[m][d] * Cs[d][col] : 4 chained f32 WMMAs over D=16.
    v8f acc = {};
#pragma unroll
    for (int c = 0; c < 4; ++c) {
      acc = __builtin_amdgcn_wmma_f32_16x16x4_f32(
          /*neg_a=*/false, a[c], /*neg_b=*/false, bmat[c],
          /*c_mod=*/(short)0, acc, /*reuse_a=*/false, /*reuse_b=*/false);
    }

    // C/D layout: VGPR v holds (M=v, lanes 0-15) and (M=v+8, lanes 16-31).
#pragma unroll
    for (int v = 0; v < 8; ++v) {
      O[(size_t)(point_base + v + 8 * half) * RBF_K + col] =
          hval[v] + csq - 2.0f * acc[v];
    }
  }
}

extern "C" void kernel_launch(void* const* d_in, const int* in_sizes, int n_in,
                              void* d_out, int out_size, void* d_ws, size_t ws_size,
                              hipStream_t stream) {
  const float* x       = (const float*)d_in[0];  // (B,N,D) f32
  const float* centers = (const float*)d_in[1];  // (K,D)   f32
  const float* scaling = (const float*)d_in[2];  // (D,)    f32

  const int npoints = in_sizes[0] / RBF_D;       // B*N = 65536
  float* O  = (float*)d_out;                     // (B,N,K) first
  float* Hx = O + (size_t)npoints * RBF_K;       // then (B,N,D)

  const int points_per_block = 16 * 8;           // 8 waves * 16 points
  const int blocks = npoints / points_per_block; // 512

  hipLaunchKernelGGL(rbf_wmma_kernel, dim3(blocks), dim3(256), 0, stream,
                     x, centers, scaling, O, Hx);
}